// RoundGIN_noparam_29257317220549
// MI455X (gfx1250) — compile-verified
//
#include <hip/hip_runtime.h>
#include <hip/hip_bf16.h>
#include <stdint.h>

// RoundGIN on MI455X (gfx1250).
// one_plus_eps == 0 -> each layer is exactly: h_next = scatter_sum(h[src] -> dst) * mask.
// Sparse gather + scatter-add; the 19.2MB per-slice working set fits the 192MB L2,
// so this is L2/atomic bound. Wave-per-edge coalesced gathers + hw f32 atomics +
// async global->LDS edge staging + prefetch look-ahead.

#define N_NODES 50000
#define N_EDGES 800000
#define D 96
#define ROW 864            // 3 windows * 3 states (h0,h1,h2) * 96
#define TILE 256           // N_EDGES % TILE == 0 -> no tail handling needed

#if defined(__has_builtin)
#  if __has_builtin(__builtin_amdgcn_global_load_async_to_lds_b32)
#    define USE_ASYNC_LDS 1
#  endif
#endif
#ifndef USE_ASYNC_LDS
#  define USE_ASYNC_LDS 0
#endif

typedef __attribute__((address_space(1))) int* gptr_i32;
typedef __attribute__((address_space(3))) int* lptr_i32;

// One aggregation layer for all 3 windows at once.
// Reads slice at column RD, atomically accumulates into slice at RD+96.
// (Disjoint columns -> no read/write hazard within the kernel.)
template <int RD>
__global__ __launch_bounds__(TILE) void rgin_edge(const int* __restrict__ src_g,
                                                  const int* __restrict__ dst_g,
                                                  float* out) {
  __shared__ int s_src[TILE];
  __shared__ int s_dst[TILE];
  const int tid = threadIdx.x;
  const int gidx = blockIdx.x * TILE + tid;

#if USE_ASYNC_LDS
  // CDNA5 async global->LDS copy of this block's edge tile (tracked by ASYNCcnt).
  __builtin_amdgcn_global_load_async_to_lds_b32(
      (gptr_i32)(uintptr_t)(src_g + gidx),
      (lptr_i32)(uintptr_t)(&s_src[tid]), 0, 0);
  __builtin_amdgcn_global_load_async_to_lds_b32(
      (gptr_i32)(uintptr_t)(dst_g + gidx),
      (lptr_i32)(uintptr_t)(&s_dst[tid]), 0, 0);
#  if __has_builtin(__builtin_amdgcn_s_wait_asynccnt)
  __builtin_amdgcn_s_wait_asynccnt(0);
#  else
  asm volatile("s_wait_asynccnt 0" ::: "memory");
#  endif
#else
  s_src[tid] = src_g[gidx];
  s_dst[tid] = dst_g[gidx];
#endif
  __syncthreads();

  const int lane = tid & 31;          // wave32
  const int wbase = tid & ~31;        // this wave's 32 staged edges

#pragma unroll 1
  for (int j = 0; j < 32; ++j) {
    const int s = s_src[wbase + j];   // LDS broadcast (uniform per wave)
    const int t = s_dst[wbase + j];

    // Prefetch next edge's source row (all 3 window segments) into cache.
    if (j + 1 < 32) {
      const int sp = s_src[wbase + j + 1];
      const char* p = (const char*)(out + (size_t)sp * ROW + RD + lane);
      __builtin_prefetch(p, 0, 3);
      __builtin_prefetch(p + 288 * 4, 0, 3);
      __builtin_prefetch(p + 576 * 4, 0, 3);
    }

    const float* srow = out + (size_t)s * ROW + RD + lane;
    float* drow = out + (size_t)t * ROW + RD + 96 + lane;

    float v[9];
#pragma unroll
    for (int w = 0; w < 3; ++w)
#pragma unroll
      for (int c = 0; c < 3; ++c)
        v[w * 3 + c] = srow[w * 288 + c * 32];   // coalesced 128B per wave-chunk

    // Masked-out source rows are all-zero across the whole wave: use a
    // wave-uniform vote per window (one s_cbranch) instead of 9 divergent
    // exec-mask toggles, then issue the atomics unconditionally.
#pragma unroll
    for (int w = 0; w < 3; ++w) {
      const float a0 = v[w * 3 + 0];
      const float a1 = v[w * 3 + 1];
      const float a2 = v[w * 3 + 2];
      const bool live = (a0 != 0.0f) || (a1 != 0.0f) || (a2 != 0.0f);
      if (__any(live)) {
        atomicAdd(&drow[w * 288 + 0],  a0);
        atomicAdd(&drow[w * 288 + 32], a1);
        atomicAdd(&drow[w * 288 + 64], a2);
      }
    }
  }
}

// Emit the edge kernels FIRST so the disasm snippet shows their codegen
// (atomic lowering + async-LDS staging).
template __global__ void rgin_edge<0>(const int*, const int*, float*);
template __global__ void rgin_edge<96>(const int*, const int*, float*);

// h0_w = feature * (age >= w); zero the h1/h2 accumulation slices.
__global__ __launch_bounds__(96) void rgin_init(const float* __restrict__ feat,
                                                const int* __restrict__ age,
                                                float* __restrict__ out) {
  const int n = blockIdx.x;
  const int d = threadIdx.x;
  const float f = feat[(size_t)n * D + d];
  const int a = age[n];
  float* row = out + (size_t)n * ROW;
#pragma unroll
  for (int w = 0; w < 3; ++w) {
    row[w * 288 + d]       = (a >= w) ? f : 0.0f;
    row[w * 288 + 96 + d]  = 0.0f;
    row[w * 288 + 192 + d] = 0.0f;
  }
}

// Apply mask_w to the freshly accumulated slice at column `off` (w=0 mask is all-ones).
__global__ __launch_bounds__(96) void rgin_mask(const int* __restrict__ age,
                                                float* __restrict__ out, int off) {
  const int n = blockIdx.x;
  const int d = threadIdx.x;
  const int a = age[n];
  float* row = out + (size_t)n * ROW;
  if (a < 1) row[1 * 288 + off + d] = 0.0f;
  if (a < 2) row[2 * 288 + off + d] = 0.0f;
}

extern "C" void kernel_launch(void* const* d_in, const int* in_sizes, int n_in,
                              void* d_out, int out_size, void* d_ws, size_t ws_size,
                              hipStream_t stream) {
  (void)in_sizes; (void)n_in; (void)out_size; (void)d_ws; (void)ws_size;
  const float* feature = (const float*)d_in[0];
  const int* age = (const int*)d_in[1];
  const int* src = (const int*)d_in[2];
  const int* dst = (const int*)d_in[3];
  float* out = (float*)d_out;

  rgin_init<<<N_NODES, 96, 0, stream>>>(feature, age, out);
  rgin_edge<0><<<N_EDGES / TILE, TILE, 0, stream>>>(src, dst, out);    // layer 1
  rgin_mask<<<N_NODES, 96, 0, stream>>>(age, out, 96);
  rgin_edge<96><<<N_EDGES / TILE, TILE, 0, stream>>>(src, dst, out);   // layer 2
  rgin_mask<<<N_NODES, 96, 0, stream>>>(age, out, 192);
}